// AVWGCN_71846212927861
// MI455X (gfx1250) — compile-verified
//
#include <hip/hip_runtime.h>
#include <hip/hip_bf16.h>

#define NB    64      // batch
#define NN    2048    // nodes
#define CD    64      // dim_in == dim_out
#define ED    10      // embed dim
#define KI    192     // CHEB_K * dim_in

typedef float v2f_t __attribute__((ext_vector_type(2)));
typedef float v8f_t __attribute__((ext_vector_type(8)));
typedef int   v4i_t __attribute__((vector_size(4 * sizeof(int))));

// ---------------------------------------------------------------------------
// Async global->LDS copy (gfx1250 GLOBAL_LOAD_ASYNC_TO_LDS_B128, ASYNCcnt).
// Probe-derived signature: (AS1 v4i* gsrc, lds dst, i32 offset, i32 cpol).
// Guarded: falls back to synchronous float4 copies if the builtin is absent.
// ---------------------------------------------------------------------------
#if defined(__gfx1250__) && __has_builtin(__builtin_amdgcn_global_load_async_to_lds_b128)
#define USE_ASYNC_LDS 1
#endif

#define AS_GLOBAL __attribute__((address_space(1)))
#define AS_SHARED __attribute__((address_space(3)))

__device__ __forceinline__ void copy16_g2l(void* lds_dst, const void* gsrc) {
#ifdef USE_ASYNC_LDS
  // Generic->AS cast via integer: LDS generic addr low 32 bits == LDS offset
  // (ISA 10.2 aperture table), global generic == AS1 numerically.
  __builtin_amdgcn_global_load_async_to_lds_b128(
      (AS_GLOBAL v4i_t*)(unsigned long long)gsrc,
      (AS_SHARED v4i_t*)(unsigned int)(unsigned long long)lds_dst,
      0, 0);
#else
  *(float4*)lds_dst = *(const float4*)gsrc;
#endif
}

__device__ __forceinline__ void wait_tile_ready() {
#ifdef USE_ASYNC_LDS
#if __has_builtin(__builtin_amdgcn_s_wait_asynccnt)
  __builtin_amdgcn_s_wait_asynccnt(0);
#else
  asm volatile("s_wait_asynccnt 0x0" ::: "memory");
#endif
#endif
  __syncthreads();
}

// ---------------------------------------------------------------------------
// Kernel 1: A = softmax(relu(E E^T)) row-wise.  One block per row.
// ---------------------------------------------------------------------------
__global__ __launch_bounds__(256) void k_supports(const float* __restrict__ E,
                                                  float* __restrict__ A) {
  const int n = blockIdx.x;
  const int t = threadIdx.x;
  __shared__ float en[16];
  __shared__ float red[256];
  if (t < ED) en[t] = E[n * ED + t];
  __syncthreads();

  float v[8];
  float mx = 0.0f;  // relu output >= 0
#pragma unroll
  for (int j = 0; j < 8; ++j) {
    const int m = t + j * 256;
    const float* em = E + m * ED;
    float d = 0.0f;
#pragma unroll
    for (int q = 0; q < ED; ++q) d += en[q] * em[q];
    d = fmaxf(d, 0.0f);
    v[j] = d;
    mx = fmaxf(mx, d);
  }
  red[t] = mx;
  __syncthreads();
  for (int s = 128; s > 0; s >>= 1) {
    if (t < s) red[t] = fmaxf(red[t], red[t + s]);
    __syncthreads();
  }
  mx = red[0];
  __syncthreads();

  float sum = 0.0f;
#pragma unroll
  for (int j = 0; j < 8; ++j) {
    v[j] = __expf(v[j] - mx);
    sum += v[j];
  }
  red[t] = sum;
  __syncthreads();
  for (int s = 128; s > 0; s >>= 1) {
    if (t < s) red[t] += red[t + s];
    __syncthreads();
  }
  const float inv = 1.0f / red[0];
#pragma unroll
  for (int j = 0; j < 8; ++j) A[(size_t)n * NN + t + j * 256] = v[j] * inv;
}

// ---------------------------------------------------------------------------
// Kernel 2: Y[b] = alpha * (Amat @ Xin[b]) + beta * Xres[b]
// Double-buffered async global->LDS staging + WMMA f32 16x16x4.
//   As: [64 rows][36]  (row stride 144 B: 16B-aligned, bank-conflict-free)
//   Xs: [32 k   ][72]  (row stride 288 B: 16B-aligned, half-waves disjoint)
// ---------------------------------------------------------------------------
#define SA 36
#define SX 72

__global__ __launch_bounds__(256) void k_gemm_AX(const float* __restrict__ Amat,
                                                 const float* __restrict__ Xin,
                                                 const float* __restrict__ Xres,
                                                 float* __restrict__ Yout,
                                                 float alpha, float beta) {
  const int b  = blockIdx.y;
  const int n0 = blockIdx.x * 64;
  const int t  = threadIdx.x;
  const int lane = t & 31;
  const int wave = t >> 5;

  __shared__ float As[2][64 * SA];
  __shared__ float Xs[2][32 * SX];

  const int rt      = wave >> 1;     // row tile 0..3
  const int c0      = wave & 1;      // col tiles c0 and c0+2
  const int halfsel = lane >> 4;     // 0 / 1
  const int rl      = lane & 15;

  const float* Xb = Xin + (size_t)b * (NN * CD);

  // chunk decomposition for the 16B copies (512 chunks per tile, 2/thread)
  const int a_row0 = t >> 3, a_c0 = (t & 7) * 4;          // A chunk for q=t
  const int a_row1 = (t + 256) >> 3, a_c1 = a_c0;          // q=t+256 (same &7)
  const int x_k0 = t >> 4, x_c0 = (t & 15) * 4;            // X chunk for q=t
  const int x_k1 = (t + 256) >> 4, x_c1 = x_c0;            // q=t+256

  v8f_t acc0 = {};
  v8f_t acc1 = {};

  // ---- prologue: stage tile 0 into buffer 0 ----
  {
    copy16_g2l(&As[0][a_row0 * SA + a_c0], Amat + (size_t)(n0 + a_row0) * NN + a_c0);
    copy16_g2l(&As[0][a_row1 * SA + a_c1], Amat + (size_t)(n0 + a_row1) * NN + a_c1);
    copy16_g2l(&Xs[0][x_k0 * SX + x_c0], Xb + (size_t)x_k0 * CD + x_c0);
    copy16_g2l(&Xs[0][x_k1 * SX + x_c1], Xb + (size_t)x_k1 * CD + x_c1);
  }
  wait_tile_ready();

  int cur = 0;
  for (int k0 = 0; k0 < NN; k0 += 32) {
    const int nxt = cur ^ 1;
    const int k1 = k0 + 32;
    if (k1 < NN) {  // issue next tile while computing on current
      copy16_g2l(&As[nxt][a_row0 * SA + a_c0], Amat + (size_t)(n0 + a_row0) * NN + k1 + a_c0);
      copy16_g2l(&As[nxt][a_row1 * SA + a_c1], Amat + (size_t)(n0 + a_row1) * NN + k1 + a_c1);
      copy16_g2l(&Xs[nxt][x_k0 * SX + x_c0], Xb + (size_t)(k1 + x_k0) * CD + x_c0);
      copy16_g2l(&Xs[nxt][x_k1 * SX + x_c1], Xb + (size_t)(k1 + x_k1) * CD + x_c1);
    }

    const float* __restrict__ Ac = As[cur];
    const float* __restrict__ Xc = Xs[cur];
#pragma unroll
    for (int kk = 0; kk < 32; kk += 4) {
      const int ka = kk + halfsel * 2;
      v2f_t a, b0, b1;
      a.x  = Ac[(rt * 16 + rl) * SA + ka];
      a.y  = Ac[(rt * 16 + rl) * SA + ka + 1];
      b0.x = Xc[ka * SX + c0 * 16 + rl];
      b0.y = Xc[(ka + 1) * SX + c0 * 16 + rl];
      b1.x = Xc[ka * SX + (c0 + 2) * 16 + rl];
      b1.y = Xc[(ka + 1) * SX + (c0 + 2) * 16 + rl];
      acc0 = __builtin_amdgcn_wmma_f32_16x16x4_f32(false, a, false, b0,
                                                   (short)0, acc0, false, false);
      acc1 = __builtin_amdgcn_wmma_f32_16x16x4_f32(false, a, false, b1,
                                                   (short)0, acc1, false, false);
    }
    wait_tile_ready();
    cur = nxt;
  }

  float* Yb = Yout + (size_t)b * (NN * CD);
  const float* Rb = Xres + (size_t)b * (NN * CD);
#pragma unroll
  for (int r = 0; r < 8; ++r) {
    const int row  = n0 + rt * 16 + r + halfsel * 8;
    const int col0 = c0 * 16 + rl;
    const int col1 = (c0 + 2) * 16 + rl;
    float y0 = alpha * acc0[r];
    float y1 = alpha * acc1[r];
    if (beta != 0.0f) {
      y0 += beta * Rb[(size_t)row * CD + col0];
      y1 += beta * Rb[(size_t)row * CD + col1];
    }
    Yb[(size_t)row * CD + col0] = y0;
    Yb[(size_t)row * CD + col1] = y1;
  }
}

// ---------------------------------------------------------------------------
// Kernel 3: per-node output.  One block per node n.
//   W_n[ki,o] = sum_d E[n,d] * Wp[d,ki,o]   (built in LDS, [192 x 73])
//   XG[b,ki]  = {x, y1, y2}[b,n,:]          (LDS, [64 x 195])
//   out[b,n,o] = sum_ki XG[b,ki] * W_n[ki,o] + bias_n[o]
// ---------------------------------------------------------------------------
#define LDW 73
#define LDX 195

__global__ __launch_bounds__(256) void k_node_out(const float* __restrict__ X,
                                                  const float* __restrict__ Y1,
                                                  const float* __restrict__ Y2,
                                                  const float* __restrict__ E,
                                                  const float* __restrict__ Wp,
                                                  const float* __restrict__ Bp,
                                                  float* __restrict__ Out) {
  const int n = blockIdx.x;
  const int t = threadIdx.x;

  extern __shared__ float smem[];
  float* Ws    = smem;                 // KI * LDW
  float* XGs   = Ws + KI * LDW;        // NB * LDX
  float* biasS = XGs + NB * LDX;       // 64
  float* es    = biasS + 64;           // 16

  if (t < ED) es[t] = E[n * ED + t];
  __syncthreads();

  float e[ED];
#pragma unroll
  for (int d = 0; d < ED; ++d) e[d] = es[d];

  // Synthesize per-node weights (float4 global reads):
  // Wp flat layout [d][k][i][o] == [d][ki*64+o]
  for (int idx4 = t * 4; idx4 < KI * CD; idx4 += 1024) {
    float ax = 0.0f, ay = 0.0f, az = 0.0f, aw = 0.0f;
#pragma unroll
    for (int d = 0; d < ED; ++d) {
      const float4 w = *(const float4*)(Wp + (size_t)d * (KI * CD) + idx4);
      ax += e[d] * w.x; ay += e[d] * w.y; az += e[d] * w.z; aw += e[d] * w.w;
    }
    const int ki = idx4 >> 6;
    const int o  = idx4 & 63;
    float* dst = Ws + ki * LDW + o;
    dst[0] = ax; dst[1] = ay; dst[2] = az; dst[3] = aw;
  }
  if (t < 64) {
    float acc = 0.0f;
#pragma unroll
    for (int d = 0; d < ED; ++d) acc += e[d] * Bp[d * CD + t];
    biasS[t] = acc;
  }
  // Gather xg rows for this node (float4): [b][k*64+c]
  for (int q = t; q < NB * (KI / 4); q += 256) {   // 3072 chunks
    const int b   = q / (KI / 4);
    const int rem = q - b * (KI / 4);
    const int k   = rem >> 4;
    const int c4  = (rem & 15) * 4;
    const float* src = (k == 0) ? X : ((k == 1) ? Y1 : Y2);
    const float4 v = *(const float4*)(src + ((size_t)b * NN + n) * CD + c4);
    float* dst = XGs + b * LDX + k * 64 + c4;
    dst[0] = v.x; dst[1] = v.y; dst[2] = v.z; dst[3] = v.w;
  }
  __syncthreads();

  const int lane = t & 31;
  const int wave = t >> 5;
  const int rt      = wave >> 1;
  const int c0      = wave & 1;
  const int halfsel = lane >> 4;
  const int rl      = lane & 15;

  v8f_t acc0 = {};
  v8f_t acc1 = {};
#pragma unroll 4
  for (int kk = 0; kk < KI; kk += 4) {
    const int ka = kk + halfsel * 2;
    v2f_t a, b0, b1;
    a.x  = XGs[(rt * 16 + rl) * LDX + ka];
    a.y  = XGs[(rt * 16 + rl) * LDX + ka + 1];
    b0.x = Ws[ka * LDW + c0 * 16 + rl];
    b0.y = Ws[(ka + 1) * LDW + c0 * 16 + rl];
    b1.x = Ws[ka * LDW + (c0 + 2) * 16 + rl];
    b1.y = Ws[(ka + 1) * LDW + (c0 + 2) * 16 + rl];
    acc0 = __builtin_amdgcn_wmma_f32_16x16x4_f32(false, a, false, b0,
                                                 (short)0, acc0, false, false);
    acc1 = __builtin_amdgcn_wmma_f32_16x16x4_f32(false, a, false, b1,
                                                 (short)0, acc1, false, false);
  }

#pragma unroll
  for (int r = 0; r < 8; ++r) {
    const int brow = rt * 16 + r + halfsel * 8;
    const int o0   = c0 * 16 + rl;
    const int o1   = (c0 + 2) * 16 + rl;
    const size_t base = ((size_t)brow * NN + n) * CD;
    Out[base + o0] = acc0[r] + biasS[o0];
    Out[base + o1] = acc1[r] + biasS[o1];
  }
}

// ---------------------------------------------------------------------------
extern "C" void kernel_launch(void* const* d_in, const int* in_sizes, int n_in,
                              void* d_out, int out_size, void* d_ws,
                              size_t ws_size, hipStream_t stream) {
  (void)in_sizes; (void)n_in; (void)out_size; (void)ws_size;
  const float* x  = (const float*)d_in[0];  // [64, 2048, 64]
  const float* E  = (const float*)d_in[1];  // [2048, 10]
  const float* Wp = (const float*)d_in[2];  // [10, 3, 64, 64]
  const float* Bp = (const float*)d_in[3];  // [10, 64]
  // d_in[4] = cheb_k (scalar, fixed at 3)

  float* A   = (float*)d_ws;                      // 2048*2048
  float* Y1  = A + (size_t)NN * NN;               // 64*2048*64
  float* Y2  = Y1 + (size_t)NB * NN * CD;         // 64*2048*64
  float* Out = (float*)d_out;

  // 1) adaptive adjacency
  k_supports<<<NN, 256, 0, stream>>>(E, A);

  // 2) Chebyshev propagation: y1 = A x ; y2 = 2 A y1 - x
  dim3 g2(NN / 64, NB);
  k_gemm_AX<<<g2, 256, 0, stream>>>(A, x, x, Y1, 1.0f, 0.0f);
  k_gemm_AX<<<g2, 256, 0, stream>>>(A, Y1, x, Y2, 2.0f, -1.0f);

  // 3) per-node weight synthesis + mixing GEMM + bias
  const size_t smem3 =
      (size_t)(KI * LDW + NB * LDX + 64 + 16) * sizeof(float);
  k_node_out<<<NN, 256, smem3, stream>>>(x, Y1, Y2, E, Wp, Bp, Out);
}